// HealEncoding_33938831573234
// MI455X (gfx1250) — compile-verified
//
#include <hip/hip_runtime.h>
#include <stdint.h>

// HealEncoding: L=10 levels, F=8 features, 4-neighbor weighted gather-sum.
// out[n, f*10 + l] = sum_k w[l,k,n] * params[STARTS[l] + idx[l,k,n], f]
// STARTS[l] = 4^(l+1) - 4 (12*4^l pixels per level, prefix sum).
//
// Memory-bound gather: params table (134 MB) fits MI455X's 192 MB L2, so the
// win is (a) NT hints on the 320 MB of single-use streams so they don't evict
// the table, (b) b128 gathers in 32-bit saddr+voffset form, (c) LDS-staged
// coalesced output emitted with CDNA5 global_store_async_from_lds_b128
// (ASYNCcnt path, NT), (d) global_prefetch of next level's rows one level ahead.

#define N_LVL 10
#define FDIM  8
#define TPB   128
#define LDS_STRIDE 84   // 80 data floats + 4 pad: 336 B (16B aligned), stride%64banks=20 -> 2-way max

typedef float v4f __attribute__((ext_vector_type(4)));
typedef int   v4i __attribute__((ext_vector_type(4)));

#if defined(__gfx1250__) && __has_builtin(__builtin_amdgcn_global_store_async_from_lds_b128)
#define USE_ASYNC_OUT 1
typedef __attribute__((address_space(1))) v4i gbl_v4i;   // global-AS int4 (builtin's dst type)
typedef __attribute__((address_space(3))) v4i lds_v4i;   // LDS-AS int4 (builtin's src type)
#else
#define USE_ASYNC_OUT 0
#endif

__global__ __launch_bounds__(TPB) void heal_enc_kernel(
    const float* __restrict__ params,
    const int*   __restrict__ nidx,   // [L,4,N]
    const float* __restrict__ nw,     // [L,4,N]
    float*       __restrict__ out,    // [N, F*L], f-major l-minor
    const int N)
{
  __shared__ float stage[TPB * LDS_STRIDE];

  const int  tid = threadIdx.x;
  const int  n0  = blockIdx.x * TPB;
  const int  n   = n0 + tid;
  const long NL  = (long)N;

  if (n < N) {
    // level-0 neighbor indices (coalesced, non-temporal: single-use stream)
    int i0 = __builtin_nontemporal_load(nidx + n);
    int i1 = __builtin_nontemporal_load(nidx + NL + n);
    int i2 = __builtin_nontemporal_load(nidx + 2 * NL + n);
    int i3 = __builtin_nontemporal_load(nidx + 3 * NL + n);
    int start = 0;                       // STARTS[0]

    float* my = &stage[tid * LDS_STRIDE];

#pragma unroll
    for (int l = 0; l < N_LVL; ++l) {
      // 4 row gathers, 32B each -> 2x global_load_b128 per row.
      // 32-bit float indices (max ~33.5M << 2^31): backend can use the
      // GVS saddr + sext(i32 voffset) form, one v_mad per row, no 64-bit adds.
      const int fo0 = (i0 + start) * FDIM;
      const int fo1 = (i1 + start) * FDIM;
      const int fo2 = (i2 + start) * FDIM;
      const int fo3 = (i3 + start) * FDIM;
      const v4f* r0 = (const v4f*)(params + fo0);
      const v4f* r1 = (const v4f*)(params + fo1);
      const v4f* r2 = (const v4f*)(params + fo2);
      const v4f* r3 = (const v4f*)(params + fo3);
      v4f a0 = r0[0], a1 = r0[1];
      v4f b0 = r1[0], b1 = r1[1];
      v4f c0 = r2[0], c1 = r2[1];
      v4f d0 = r3[0], d1 = r3[1];

      const float* wl = nw + ((long)l * 4) * NL + n;
      float w0 = __builtin_nontemporal_load(wl);
      float w1 = __builtin_nontemporal_load(wl + NL);
      float w2 = __builtin_nontemporal_load(wl + 2 * NL);
      float w3 = __builtin_nontemporal_load(wl + 3 * NL);

      // software pipeline: pull next level's indices now, prefetch its rows
      // (global_prefetch_b8) while this level's gathers are still in flight.
      const int start_next = (4 << (2 * (l + 1))) - 4;   // STARTS[l+1]
      if (l + 1 < N_LVL) {
        const int* ipn = nidx + ((long)(l + 1) * 4) * NL + n;
        int j0 = __builtin_nontemporal_load(ipn);
        int j1 = __builtin_nontemporal_load(ipn + NL);
        int j2 = __builtin_nontemporal_load(ipn + 2 * NL);
        int j3 = __builtin_nontemporal_load(ipn + 3 * NL);
        __builtin_prefetch(params + (j0 + start_next) * FDIM);
        __builtin_prefetch(params + (j1 + start_next) * FDIM);
        __builtin_prefetch(params + (j2 + start_next) * FDIM);
        __builtin_prefetch(params + (j3 + start_next) * FDIM);
        i0 = j0; i1 = j1; i2 = j2; i3 = j3;
      }

      v4f lo = w0 * a0 + w1 * b0 + w2 * c0 + w3 * d0;   // f = 0..3
      v4f hi = w0 * a1 + w1 * b1 + w2 * c1 + w3 * d1;   // f = 4..7

      // stage in the final f-major/l-minor layout; across the unrolled levels
      // each f's 10 values are contiguous, so the backend merges these into
      // wide ds_store's.
      my[0 * N_LVL + l] = lo.x;
      my[1 * N_LVL + l] = lo.y;
      my[2 * N_LVL + l] = lo.z;
      my[3 * N_LVL + l] = lo.w;
      my[4 * N_LVL + l] = hi.x;
      my[5 * N_LVL + l] = hi.y;
      my[6 * N_LVL + l] = hi.z;
      my[7 * N_LVL + l] = hi.w;

      start = start_next;
    }
  }

  __syncthreads();

  // Block's output region is one contiguous [pts*80] float run -> fully
  // coalesced b128 emission straight out of LDS.
  int pts = N - n0;
  if (pts > TPB) pts = TPB;
  const int totalV = pts * (FDIM * N_LVL / 4);          // 20 x float4 per point
  float* outBase = out + (long)n0 * (FDIM * N_LVL);

#if USE_ASYNC_OUT
  for (int q = tid; q < totalV; q += TPB) {
    const int point = q / 20;
    const int offF  = (q - point * 20) * 4;             // multiple of 4 floats
    float* g = outBase + (long)point * (FDIM * N_LVL) + offF;
    float* s = &stage[point * LDS_STRIDE + offF];
    // cpol=1 -> TH=NT: 160 MB single-use output must not evict params from L2
    __builtin_amdgcn_global_store_async_from_lds_b128(
        (gbl_v4i*)g, (lds_v4i*)s, /*imm offset*/0, /*cpol NT*/1);
  }
#if __has_builtin(__builtin_amdgcn_s_wait_asynccnt)
  __builtin_amdgcn_s_wait_asynccnt(0);
#endif
#else
  for (int q = tid; q < totalV; q += TPB) {
    const int point = q / 20;
    const int offF  = (q - point * 20) * 4;
    v4f v = *(const v4f*)&stage[point * LDS_STRIDE + offF];
    __builtin_nontemporal_store(v, (v4f*)(outBase + (long)point * (FDIM * N_LVL) + offF));
  }
#endif
}

extern "C" void kernel_launch(void* const* d_in, const int* in_sizes, int n_in,
                              void* d_out, int out_size, void* d_ws, size_t ws_size,
                              hipStream_t stream) {
  const float* params = (const float*)d_in[0];
  const int*   nidx   = (const int*)d_in[1];
  const float* nw     = (const float*)d_in[2];
  float*       out    = (float*)d_out;

  const int N = in_sizes[1] / (N_LVL * 4);   // neigh_pix is [L,4,N]
  const int blocks = (N + TPB - 1) / TPB;
  heal_enc_kernel<<<blocks, TPB, 0, stream>>>(params, nidx, nw, out, N);
}